// Mamba2Block_43224550867455
// MI455X (gfx1250) — compile-verified
//
#include <hip/hip_runtime.h>
#include <hip/hip_bf16.h>

typedef __attribute__((ext_vector_type(2))) float v2f;
typedef __attribute__((ext_vector_type(8))) float v8f;
typedef __attribute__((ext_vector_type(4))) unsigned int u32x4;
typedef __attribute__((ext_vector_type(8))) int i32x8;
typedef __attribute__((ext_vector_type(4))) int i32x4;

#define DT_MIN 0.001f
#define DT_MAX 0.1f

// Flat pointer -> 32-bit LDS address (ISA 10.2: for the LDS aperture the
// hardware uses addr[31:0] as the LDS address).
__device__ __forceinline__ unsigned lds_addr_of(const void* p) {
  return (unsigned)(unsigned long long)(uintptr_t)p;
}

// ---------------------------------------------------------------------------
// Issue one 2D Tensor-Data-Mover tile load (TENSOR_LOAD_TO_LDS).
// Tile = tile_d1 rows x tile_d0 elements from a row-major f32 tensor whose
// row stride is stride_d0 elements; destination LDS at `lds`, with
// (pad_amt_m1+1) dwords of padding inserted after every 2^(pad_iv+1) dwords
// (ISA 8.4: D# group1 pad fields) so the LDS image lands bank-padded.
// Descriptor layout per CDNA5 ISA 8.3/8.4 (group0 128b, group1 256b,
// groups 2/3 zero => 2D tensor). This toolchain exposes the 6-arg builtin
// (g0, g1, g2, g3, reserved i32x8, cpol).
// ---------------------------------------------------------------------------
__device__ __forceinline__ void tdm_load_tile_2d(
    unsigned lds, const float* gptr,
    unsigned tensor_d0, unsigned tensor_d1, unsigned long long stride_d0,
    unsigned tile_d0, unsigned tile_d1, unsigned pad_iv, unsigned pad_amt_m1) {
  const unsigned long long ga = (unsigned long long)(uintptr_t)gptr;
  u32x4 g0;
  g0[0] = 1u;                                                  // count=1, user D#
  g0[1] = lds;                                                 // lds_addr
  g0[2] = (unsigned)ga;                                        // global_addr[31:0]
  g0[3] = (unsigned)((ga >> 32) & 0x01FFFFFFu) | (2u << 30);   // addr[56:32] | type=2
  i32x8 g1;
  g1[0] = (int)((2u << 16)              // data_size = 4 bytes
              | (1u << 20)              // pad_enable
              | (pad_iv << 22)          // pad interval code (2^(v+1) dwords)
              | (pad_amt_m1 << 25));    // pad amount - 1 (dwords)
  g1[1] = (int)((tensor_d0 & 0xFFFFu) << 16);                              // dim0 lo
  g1[2] = (int)(((tensor_d0 >> 16) & 0xFFFFu) | ((tensor_d1 & 0xFFFFu) << 16));
  g1[3] = (int)(((tensor_d1 >> 16) & 0xFFFFu) | ((tile_d0 & 0xFFFFu) << 16));
  g1[4] = (int)(tile_d1 & 0xFFFFu);                                        // tile_dim2=0
  g1[5] = (int)(unsigned)(stride_d0 & 0xFFFFFFFFu);
  g1[6] = (int)(unsigned)((stride_d0 >> 32) & 0xFFFFu);                    // dim1_stride=0
  g1[7] = 0;
  const i32x4 gz4 = {0, 0, 0, 0};
  const i32x8 gz8 = {0, 0, 0, 0, 0, 0, 0, 0};
  __builtin_amdgcn_tensor_load_to_lds(g0, g1, gz4, gz4, gz8, 0);
}

// ---------------------------------------------------------------------------
// WMMA f32 GEMM: out[M x Nt] = A[M x K] * W[K x Nt], row-major.
// 128 threads (4 waves), block tile 64x64, K-stage 16, TDM double-buffered.
// Wave w owns rows [w*16, w*16+16) x 64 cols as 4 C fragments.
// MODE 0: plain store
// MODE 1: split columns at Esplit -> out0 / out1
// MODE 2: v = clip(softplus(v + bias[col]), DT_MIN, DT_MAX)
// ---------------------------------------------------------------------------
template <int MODE>
__global__ __launch_bounds__(128)
void gemm_wmma_f32(const float* __restrict__ A, const float* __restrict__ W,
                   float* __restrict__ out0, float* __restrict__ out1,
                   const float* __restrict__ bias,
                   int M, int K, int Nt, int Esplit) {
  // TDM pad gives row strides 17 (A) and 72 (B): 17 is coprime with 64 banks;
  // 2*72 mod 64 == 16 keeps the two half-wave K-rows on disjoint banks.
  __shared__ float As[2][64][17];
  __shared__ float Bs[2][16][72];

  const int tid  = threadIdx.x;
  const int lane = tid & 31;
  const int wave = tid >> 5;         // 0..3 -> M sub-tile
  const int l16  = lane & 15;
  const int kh   = (lane >> 4) * 2;  // 0 or 2: this lane's K pair

  const int blockM = blockIdx.y * 64;
  const int blockN = blockIdx.x * 64;
  const bool issuer = (tid < 32);    // wave 0 drives the TDM (per-wave op)

  if (issuer) {
    tdm_load_tile_2d(lds_addr_of(&As[0][0][0]), &A[(size_t)blockM * K],
                     (unsigned)K, (unsigned)M, (unsigned long long)K,
                     16u, 64u, /*pad 1 dw / 16 dw*/ 3u, 0u);
    tdm_load_tile_2d(lds_addr_of(&Bs[0][0][0]), &W[blockN],
                     (unsigned)Nt, (unsigned)K, (unsigned long long)Nt,
                     64u, 16u, /*pad 8 dw / 64 dw*/ 5u, 7u);
  }

  v8f acc[4] = {};
  int buf = 0;

  for (int kb = 0; kb < K; kb += 16) {
    if (issuer) __builtin_amdgcn_s_wait_tensorcnt(0);  // stage kb resident
    __syncthreads();  // everyone sees the tiles; prior-parity reads all done
    if (issuer && (kb + 16) < K) {  // launch next stage into the other buffer
      tdm_load_tile_2d(lds_addr_of(&As[buf ^ 1][0][0]),
                       &A[(size_t)blockM * K + kb + 16],
                       (unsigned)K, (unsigned)M, (unsigned long long)K,
                       16u, 64u, 3u, 0u);
      tdm_load_tile_2d(lds_addr_of(&Bs[buf ^ 1][0][0]),
                       &W[(size_t)(kb + 16) * Nt + blockN],
                       (unsigned)Nt, (unsigned)K, (unsigned long long)Nt,
                       64u, 16u, 5u, 7u);
    }

    #pragma unroll
    for (int kk = 0; kk < 16; kk += 4) {
      // A fragment 16x4: lanes 0-15 hold K={kk,kk+1}, lanes 16-31 K={kk+2,kk+3}
      v2f a;
      a.x = As[buf][wave * 16 + l16][kk + kh];
      a.y = As[buf][wave * 16 + l16][kk + kh + 1];
      #pragma unroll
      for (int f = 0; f < 4; ++f) {
        // B fragment 4x16, mirrored K halves across the wave
        v2f b;
        b.x = Bs[buf][kk + kh][f * 16 + l16];
        b.y = Bs[buf][kk + kh + 1][f * 16 + l16];
        acc[f] = __builtin_amdgcn_wmma_f32_16x16x4_f32(
            /*neg_a=*/false, a, /*neg_b=*/false, b,
            /*c_mod=*/(short)0, acc[f], /*reuse_a=*/false, /*reuse_b=*/false);
      }
    }
    buf ^= 1;
  }

  // Epilogue. C/D layout: VGPR j -> row (j + 8*(lane>=16)), col = lane&15.
  const int baseRow = blockM + wave * 16 + ((lane >> 4) << 3);
  #pragma unroll
  for (int f = 0; f < 4; ++f) {
    const int col = blockN + f * 16 + l16;
    #pragma unroll
    for (int j = 0; j < 8; ++j) {
      float v = acc[f][j];
      const int row = baseRow + j;
      if (MODE == 2) {
        v += bias[col];
        v = (v > 20.0f) ? v : log1pf(__expf(v));
        v = fminf(fmaxf(v, DT_MIN), DT_MAX);
        out0[(size_t)row * Nt + col] = v;
      } else if (MODE == 1) {
        if (col < Esplit) out0[(size_t)row * Esplit + col] = v;
        else              out1[(size_t)row * Esplit + (col - Esplit)] = v;
      } else {
        out0[(size_t)row * Nt + col] = v;
      }
    }
  }
}

// ---------------------------------------------------------------------------
// Depthwise conv over T (SAME, K=4 => pad_left=1; XLA cross-correlation:
// out[t] = sum_k w[k] * in[t+k-1]) + SiLU. Memory-bound elementwise.
// ---------------------------------------------------------------------------
__global__ __launch_bounds__(256)
void conv_silu_kernel(const float* __restrict__ xz, const float* __restrict__ Wc,
                      float* __restrict__ xc, int Bn, int T, int E) {
  const size_t idx = (size_t)blockIdx.x * blockDim.x + threadIdx.x;
  const size_t total = (size_t)Bn * T * E;
  if (idx >= total) return;
  const int e = (int)(idx % E);
  const int t = (int)((idx / E) % T);
  const int b = (int)(idx / ((size_t)E * T));
  float s = 0.0f;
  #pragma unroll
  for (int k = 0; k < 4; ++k) {
    const int tt = t + k - 1;
    if (tt >= 0 && tt < T)
      s += Wc[k * E + e] * xz[((size_t)b * T + tt) * E + e];
  }
  xc[idx] = s / (1.0f + __expf(-s));  // silu
}

// ---------------------------------------------------------------------------
// SSM selective scan: one thread per (b,e), h[64] in VGPRs; per-timestep
// B_t / C_t and -exp(A_log) staged in LDS for the whole block.
// ---------------------------------------------------------------------------
__global__ __launch_bounds__(256)
void ssm_scan_kernel(const float* __restrict__ xc, const float* __restrict__ dt,
                     const float* __restrict__ Bp, const float* __restrict__ Cp,
                     const float* __restrict__ A_log, float* __restrict__ y,
                     int T, int E) {
  __shared__ float Asm[64];
  __shared__ float Bsm[64];
  __shared__ float Csm[64];
  const int tid = threadIdx.x;
  const int b   = blockIdx.y;
  const int e   = blockIdx.x * 256 + tid;

  if (tid < 64) Asm[tid] = -__expf(A_log[tid]);
  __syncthreads();

  float h[64];
  #pragma unroll
  for (int n = 0; n < 64; ++n) h[n] = 0.0f;

  const size_t baseBT = (size_t)b * T;
  for (int t = 0; t < T; ++t) {
    if (tid < 64)       Bsm[tid]      = Bp[(baseBT + t) * 64 + tid];
    else if (tid < 128) Csm[tid - 64] = Cp[(baseBT + t) * 64 + (tid - 64)];
    __syncthreads();

    const size_t o  = (baseBT + t) * E + e;
    const float xv  = xc[o];
    const float dtv = dt[o];
    float acc = 0.0f;
    #pragma unroll
    for (int n = 0; n < 64; ++n) {
      const float dA = __expf(dtv * Asm[n]);
      h[n] = fmaf(dA, h[n], dtv * Bsm[n] * xv);
      acc  = fmaf(h[n], Csm[n], acc);
    }
    y[o] = acc;
    __syncthreads();
  }
}

// y *= silu(z), elementwise in place.
__global__ __launch_bounds__(256)
void gate_kernel(float* __restrict__ y, const float* __restrict__ z, size_t n) {
  const size_t i = (size_t)blockIdx.x * blockDim.x + threadIdx.x;
  if (i < n) {
    const float zv = z[i];
    y[i] *= zv / (1.0f + __expf(-zv));
  }
}

extern "C" void kernel_launch(void* const* d_in, const int* in_sizes, int n_in,
                              void* d_out, int out_size, void* d_ws, size_t ws_size,
                              hipStream_t stream) {
  const float* x      = (const float*)d_in[0];
  const float* W_in   = (const float*)d_in[1];
  const float* W_conv = (const float*)d_in[2];
  const float* W_dt   = (const float*)d_in[3];
  const float* b_dt   = (const float*)d_in[4];
  const float* W_B    = (const float*)d_in[5];
  const float* W_C    = (const float*)d_in[6];
  const float* A_log  = (const float*)d_in[7];
  const float* W_out  = (const float*)d_in[8];
  float* out = (float*)d_out;

  const int Bn = 4, T = 2048, D = 1024, E = 2048, Nst = 64;
  const int M = Bn * T;  // 8192

  float* ws = (float*)d_ws;
  size_t off = 0;
  float* xz  = ws + off; off += (size_t)M * E;   // x_z, later reused for y / y_gated
  float* z   = ws + off; off += (size_t)M * E;
  float* xc  = ws + off; off += (size_t)M * E;
  float* dtb = ws + off; off += (size_t)M * E;
  float* Bp  = ws + off; off += (size_t)M * Nst;
  float* Cp  = ws + off; off += (size_t)M * Nst;

  const dim3 blk(128);

  // 1) xp = x @ W_in, split into x_z | z
  gemm_wmma_f32<1><<<dim3((2 * E) / 64, M / 64), blk, 0, stream>>>(
      x, W_in, xz, z, nullptr, M, D, 2 * E, E);

  // 2) depthwise conv + silu
  {
    const size_t n = (size_t)M * E;
    conv_silu_kernel<<<(unsigned)((n + 255) / 256), 256, 0, stream>>>(xz, W_conv, xc, Bn, T, E);
  }

  // 3) dt = clip(softplus(x_conv @ W_dt + b_dt))
  gemm_wmma_f32<2><<<dim3(E / 64, M / 64), blk, 0, stream>>>(
      xc, W_dt, dtb, nullptr, b_dt, M, E, E, 0);

  // 4) B_param, C_param (Nt = 64 -> one block column)
  gemm_wmma_f32<0><<<dim3(1, M / 64), blk, 0, stream>>>(
      xc, W_B, Bp, nullptr, nullptr, M, E, Nst, 0);
  gemm_wmma_f32<0><<<dim3(1, M / 64), blk, 0, stream>>>(
      xc, W_C, Cp, nullptr, nullptr, M, E, Nst, 0);

  // 5) sequential SSM scan -> y (reuse xz buffer)
  ssm_scan_kernel<<<dim3(E / 256, Bn), 256, 0, stream>>>(xc, dtb, Bp, Cp, A_log, xz, T, E);

  // 6) y *= silu(z)
  {
    const size_t n = (size_t)M * E;
    gate_kernel<<<(unsigned)((n + 255) / 256), 256, 0, stream>>>(xz, z, n);
  }

  // 7) out = y @ W_out
  gemm_wmma_f32<0><<<dim3(D / 64, M / 64), blk, 0, stream>>>(
      xz, W_out, out, nullptr, nullptr, M, E, D, 0);
}